// mesh_with_attributes_59931973648992
// MI455X (gfx1250) — compile-verified
//
#include <hip/hip_runtime.h>

typedef float v2f __attribute__((ext_vector_type(2)));
typedef float v8f __attribute__((ext_vector_type(8)));

// ---------------------------------------------------------------------------
// Kernel 1: one wave (32 lanes) per pixel.
//   lane = channel c (0..31) for the vertex-attr blend (coalesced 128B row
//   gathers from attr_per_vertex[tid,k,:]), float2 store packs both layers ->
//   fully coalesced 256B stores per wave.
//   Face attrs: lanes 0..15 -> layer 0, lanes 16..31 -> layer 1 (64B row each).
// ---------------------------------------------------------------------------
__global__ void attr_kernel(const float* __restrict__ attr,      // [T,3,32]
                            const int*   __restrict__ fid,       // [H,W,2]
                            const float* __restrict__ mask,      // [H,W,1,2]
                            const float* __restrict__ bary,      // [H,W,3,2]
                            const float* __restrict__ fragAttr,  // [T,16]
                            float* __restrict__ outRVA,          // [H,W,32,2]
                            float* __restrict__ outRFA,          // [H,W,16,2]
                            int HW) {
  const int lane = threadIdx.x & 31;
  const int wave = threadIdx.x >> 5;
  const int p = blockIdx.x * (blockDim.x >> 5) + wave;   // wave-uniform pixel id
  if (p >= HW) return;                                   // wave-uniform exit

  const int t0 = fid[2 * p + 0];
  const int t1 = fid[2 * p + 1];

  // barycentrics laid out [.., k, l]: flat = (p*3 + k)*2 + l
  const float b00 = bary[6 * p + 0], b01 = bary[6 * p + 1];
  const float b10 = bary[6 * p + 2], b11 = bary[6 * p + 3];
  const float b20 = bary[6 * p + 4], b21 = bary[6 * p + 5];
  const float m0 = mask[2 * p + 0];
  const float m1 = mask[2 * p + 1];

  // --- vertex attribute blend (channel = lane) ---
  const float* a0 = attr + (size_t)t0 * 96;
  const float* a1 = attr + (size_t)t1 * 96;
  const float acc0 = b00 * a0[lane] + b10 * a0[32 + lane] + b20 * a0[64 + lane];
  const float acc1 = b01 * a1[lane] + b11 * a1[32 + lane] + b21 * a1[64 + lane];
  float2 v = make_float2(acc0, acc1);
  *reinterpret_cast<float2*>(outRVA + (size_t)p * 64 + (size_t)lane * 2) = v;

  // --- face attribute * mask: half-wave per layer ---
  const int   f  = lane & 15;
  const int   ls = lane >> 4;            // layer select
  const int   tt = ls ? t1 : t0;
  const float mm = ls ? m1 : m0;
  outRFA[(size_t)p * 32 + (size_t)f * 2 + ls] =
      fragAttr[(size_t)tt * 16 + f] * mm;
}

// ---------------------------------------------------------------------------
// Kernel 2: one wave per 16 pixel-layers. The uniform 4x4 camera transform is
// done with V_WMMA_F32_16X16X4_F32 on the *transposed* product:
//   D(16x16) = A(16x4) x B(4x16),  A = cam_w2c_4x4_T^T (rows 4..15 zero),
//   B[:, n]  = global_pos of pixel-layer n (barycentric blend of tri pos).
// Then D[2] on lane n = z_n and D[3] on lane n = w_n -> per-lane divide,
// no cross-lane extraction needed.
// A/B per ISA layout: VGPR0 holds K={0,2}, VGPR1 holds K={1,3}; lanes 0..15
// are the K=0/1 half, lanes 16..31 the K=2/3 half.
// ---------------------------------------------------------------------------
__global__ void depth_kernel(const int*   __restrict__ fid,     // [H,W,2]
                             const float* __restrict__ mask,    // [H,W,1,2]
                             const float* __restrict__ bary,    // [H,W,3,2]
                             const float* __restrict__ cam,     // [4,4] row-major
                             const float* __restrict__ tri,     // [T,3,4]
                             float* __restrict__ outDepth,      // [H,W,1,2]
                             float* __restrict__ outMask,       // [H,W,1,2]
                             int HWL) {
  const int lane = threadIdx.x & 31;
  const int wave = threadIdx.x >> 5;
  const int base = (blockIdx.x * (blockDim.x >> 5) + wave) * 16; // wave-uniform
  if (base >= HWL) return;                                       // wave-uniform

  const int m  = lane & 15;       // pixel-layer within the 16-wide tile
  const int hi = lane >> 4;       // 0 -> components {0,1}, 1 -> {2,3}
  const int g  = base + m;        // global pixel-layer index
  const int p  = g >> 1;
  const int l  = g & 1;

  // gather triangle positions for "my" two homogeneous components
  const int tid = fid[g];
  const float b0 = bary[(size_t)(p * 3 + 0) * 2 + l];
  const float b1 = bary[(size_t)(p * 3 + 1) * 2 + l];
  const float b2 = bary[(size_t)(p * 3 + 2) * 2 + l];
  const float* tp = tri + (size_t)tid * 12 + (size_t)hi * 2;   // 8B aligned
  const float2 r0 = *reinterpret_cast<const float2*>(tp + 0);
  const float2 r1 = *reinterpret_cast<const float2*>(tp + 4);
  const float2 r2 = *reinterpret_cast<const float2*>(tp + 8);

  v2f B;  // B[K=hi*2+0][N=m], B[K=hi*2+1][N=m] = global_pos components
  B.x = b0 * r0.x + b1 * r1.x + b2 * r2.x;
  B.y = b0 * r0.y + b1 * r1.y + b2 * r2.y;

  v2f A;  // A[M=m][K=hi*2+{0,1}] = cam_w2c_4x4_T[K][m], zero-padded rows m>=4
  const bool valid = (m < 4);
  A.x = valid ? cam[(hi * 2 + 0) * 4 + m] : 0.0f;
  A.y = valid ? cam[(hi * 2 + 1) * 4 + m] : 0.0f;

  v8f C = {};
  v8f D = __builtin_amdgcn_wmma_f32_16x16x4_f32(
      /*neg_a=*/false, A, /*neg_b=*/false, B,
      /*c_mod=*/(short)0, C, /*reuse_a=*/false, /*reuse_b=*/false);

  // lane n (<16): D[2] = cam_z of pixel-layer n, D[3] = cam_w
  const float z  = D[2];
  const float w  = D[3];
  const float mk = mask[g];
  const float depth = (z / (w + 1e-6f)) * mk + (mk - 1.0f);

  if (lane < 16) {              // divergence after the WMMA is fine
    outDepth[g] = depth;
    outMask[g]  = mk;
  }
}

// ---------------------------------------------------------------------------
// Host launch
// ---------------------------------------------------------------------------
extern "C" void kernel_launch(void* const* d_in, const int* in_sizes, int n_in,
                              void* d_out, int out_size, void* d_ws, size_t ws_size,
                              hipStream_t stream) {
  const float* attr     = (const float*)d_in[0];   // [T,3,32]
  const int*   fid      = (const int*)  d_in[1];   // [H,W,2] (int32)
  const float* mask     = (const float*)d_in[2];   // [H,W,1,2]
  const float* bary     = (const float*)d_in[3];   // [H,W,3,2]
  const float* cam      = (const float*)d_in[4];   // [4,4]
  const float* tri      = (const float*)d_in[5];   // [T,3,4]
  const float* fragAttr = (const float*)d_in[6];   // [T,16]

  const int HWL = in_sizes[1];      // H*W*L
  const int HW  = HWL / 2;

  float* out      = (float*)d_out;
  float* outRVA   = out;                                  // H*W*32*2
  float* outRFA   = outRVA + (size_t)HW * 64;             // H*W*16*2
  float* outDepth = outRFA + (size_t)HW * 32;             // H*W*2
  float* outMask  = outDepth + (size_t)HW * 2;            // H*W*2

  {
    const int wavesPerBlock = 8;                           // 256 threads
    const int blocks = (HW + wavesPerBlock - 1) / wavesPerBlock;
    attr_kernel<<<blocks, 256, 0, stream>>>(attr, fid, mask, bary, fragAttr,
                                            outRVA, outRFA, HW);
  }
  {
    const int tiles = (HWL + 15) / 16;                     // 16 pixel-layers/wave
    const int blocks = (tiles + 7) / 8;                    // 8 waves/block
    depth_kernel<<<blocks, 256, 0, stream>>>(fid, mask, bary, cam, tri,
                                             outDepth, outMask, HWL);
  }
}